// EvalEig_74844100100630
// MI455X (gfx1250) — compile-verified
//
#include <hip/hip_runtime.h>
#include <hip/hip_bf16.h>

// Problem constants (must match the reference)
#define RN       1000      // matrix dimension / radial grid points
#define LMAXP1   3         // l in {0,1,2}
#define BDIM     8         // batch
#define NTHREADS 1024      // 32 waves of wave32 per block
#define PARA0f   1.0f

// gfx1250 async global->LDS path, guarded so either toolchain compiles.
#if defined(__has_builtin)
#  if __has_builtin(__builtin_amdgcn_global_load_async_to_lds_b32)
#    define HAVE_ASYNC_LDS 1
#  endif
#  if __has_builtin(__builtin_amdgcn_s_wait_asynccnt)
#    define HAVE_WAIT_ASYNC_BUILTIN 1
#  endif
#endif

#define AS1 __attribute__((address_space(1)))
#define AS3 __attribute__((address_space(3)))

__device__ __forceinline__ void wait_asynccnt0() {
#if defined(HAVE_WAIT_ASYNC_BUILTIN)
  __builtin_amdgcn_s_wait_asynccnt(0);
#else
  asm volatile("s_wait_asynccnt 0" ::: "memory");
#endif
}

// One block per (b,l) matrix. Each lane k < RN bisects for eigenvalue k of the
// symmetric tridiagonal matrix:
//   diag[i]   = 2*PARA0/RN^2 + ptl[b][i] + PARA0 * l(l+1) / r_i^2,  r_i=(i+1)/RN
//   offdiag   = -PARA0/RN^2  (constant)
// Sturm count via LDL recurrence: d = (a_i - x) - e^2 / d ; count sign(d)<0.
__global__ __launch_bounds__(NTHREADS)
void eig_tridiag_bisect(const float* __restrict__ ptl, float* __restrict__ out) {
  __shared__ __align__(16) float s_diag[RN];   // tridiagonal main diagonal
  __shared__ float s_ptl[RN];                  // staged potential row
  __shared__ float s_red[NTHREADS];            // min/max reduction scratch
  __shared__ float s_bounds[2];                // [lo, hi] Gershgorin interval

  const int bl = blockIdx.x;                   // = b*LMAXP1 + l
  const int l  = bl % LMAXP1;
  const int b  = bl / LMAXP1;
  const int t  = threadIdx.x;

  const float inv_rn2 = 1.0f / ((float)RN * (float)RN);  // 1e-6
  const float offd    = -PARA0f * inv_rn2;               // e  (negative)
  const float b2      = offd * offd;                     // e^2

  const float* __restrict__ ptl_row = ptl + (size_t)b * RN;

  // ---- Stage ptl[b, :] into LDS (gfx1250 async path when available) ----
#if defined(HAVE_ASYNC_LDS)
  if (t < RN) {
    __builtin_amdgcn_global_load_async_to_lds_b32(
        (AS1 int*)(ptl_row + t), (AS3 int*)(s_ptl + t),
        /*imm_offset=*/0, /*cpol=*/0);
  }
  wait_asynccnt0();   // per-wave: async writes to LDS landed
  __syncthreads();    // cross-wave visibility
#else
  if (t < RN) s_ptl[t] = ptl_row[t];
  __syncthreads();
#endif

  // ---- Build the diagonal in LDS + seed the min-reduction ----
  const float ll1 = (float)(l * (l + 1));
  if (t < RN) {
    const float r = (float)(t + 1) * (1.0f / (float)RN);
    const float a = 2.0f * PARA0f * inv_rn2 + s_ptl[t] + PARA0f * ll1 / (r * r);
    s_diag[t] = a;
    s_red[t]  = a;
  } else {
    s_red[t] = 3.0e38f;   // +inf surrogate for min-reduce
  }
  __syncthreads();

  // min over diag
  for (int s = NTHREADS / 2; s > 0; s >>= 1) {
    if (t < s) s_red[t] = fminf(s_red[t], s_red[t + s]);
    __syncthreads();
  }
  if (t == 0) s_bounds[0] = s_red[0];
  __syncthreads();

  // max over diag
  s_red[t] = (t < RN) ? s_diag[t] : -3.0e38f;
  __syncthreads();
  for (int s = NTHREADS / 2; s > 0; s >>= 1) {
    if (t < s) s_red[t] = fmaxf(s_red[t], s_red[t + s]);
    __syncthreads();
  }
  if (t == 0) s_bounds[1] = s_red[0];
  __syncthreads();

  if (t >= RN) return;

  // ---- Gershgorin interval (offd < 0, so min + 2*offd widens downward) ----
  float lo = s_bounds[0] + 2.0f * offd;
  float hi = s_bounds[1] - 2.0f * offd;
  lo -= 1.0e-6f + 1.0e-6f * fabsf(lo);
  hi += 1.0e-6f + 1.0e-6f * fabsf(hi);

  const int k = t;  // eigenvalue index (ascending)
  const float4* __restrict__ a4 = (const float4*)s_diag;

  #pragma unroll 1
  for (int it = 0; it < 60; ++it) {
    const float x = 0.5f * (lo + hi);
    if (!(x > lo && x < hi)) break;  // interval stagnated at fp32 resolution

    // Sturm count: number of eigenvalues < x.
    // d seeded huge so the first b2/d contribution flushes to ~0.
    int cnt = 0;
    float d = 3.0e37f;
    #pragma unroll 2
    for (int j = 0; j < RN / 4; ++j) {
      const float4 a = a4[j];   // ds_load_b128, wave-broadcast (same addr all lanes)
      d = (a.x - x) - b2 * __builtin_amdgcn_rcpf(d); cnt += (d < 0.0f);
      d = (a.y - x) - b2 * __builtin_amdgcn_rcpf(d); cnt += (d < 0.0f);
      d = (a.z - x) - b2 * __builtin_amdgcn_rcpf(d); cnt += (d < 0.0f);
      d = (a.w - x) - b2 * __builtin_amdgcn_rcpf(d); cnt += (d < 0.0f);
    }
    if (cnt > k) hi = x; else lo = x;   // maintain count(lo)<=k<count(hi)
  }

  out[(size_t)bl * RN + k] = 0.5f * (lo + hi);
}

extern "C" void kernel_launch(void* const* d_in, const int* in_sizes, int n_in,
                              void* d_out, int out_size, void* d_ws, size_t ws_size,
                              hipStream_t stream) {
  (void)in_sizes; (void)n_in; (void)d_ws; (void)ws_size; (void)out_size;
  const float* ptl = (const float*)d_in[0];   // [B, RN] float32
  float* out = (float*)d_out;                 // [B, L, RN] float32
  eig_tridiag_bisect<<<dim3(BDIM * LMAXP1), dim3(NTHREADS), 0, stream>>>(ptl, out);
}